// PeriodicConvolutionPrep_21191368639123
// MI455X (gfx1250) — compile-verified
//
#include <hip/hip_runtime.h>

typedef __attribute__((ext_vector_type(2))) float v2f;
typedef __attribute__((ext_vector_type(8))) float v8f;

#define NPTS  40000
#define DEG   16
#define CIN   16
#define COUT  16
#define GCOLS 48   // 3 * COUT

// ---------------------------------------------------------------------------
// Phase 1: G[p, d*16+o] = sum_i W[d,o,i] * F[p,i]
// Dense [NPTS x 16] @ [16 x 48] GEMM on V_WMMA_F32_16X16X4_F32.
// One wave handles one 16-row tile; K=16 is 4 chained 16x16x4 WMMAs,
// 3 N-tiles (d = 0..2) reuse the same A operand.
//
// A operand (32-bit A 16x4, ISA 7.12.2): lane L -> M = L%16,
//   VGPR0 = K = 4s + 2*(L/16), VGPR1 = K+1  -> one float2 load per K-step.
// B operand (4x16): lane L -> N = L%16, rows K = 4s + 2*(L/16), +1.
//   Wt[k, d*16+o] = W[d,o,k]; W is [3][16][16] row-major so (k,k+1) are
//   consecutive in memory -> one float2 load.
// ---------------------------------------------------------------------------
__global__ __launch_bounds__(128) void pconv_point_transform(
    const float* __restrict__ F, const float* __restrict__ W,
    float* __restrict__ G)
{
  const int lane = threadIdx.x & 31;
  const int wave = threadIdx.x >> 5;
  const int tile = blockIdx.x * 4 + wave;      // 2500 tiles of 16 points
  const int m    = lane & 15;
  const int half = lane >> 4;                  // 0: lanes 0-15, 1: lanes 16-31
  const int row  = tile * 16 + m;

  v2f a[4];
  #pragma unroll
  for (int s = 0; s < 4; ++s)
    a[s] = *(const v2f*)(F + row * CIN + 4 * s + 2 * half);

  v8f c[3] = {};
  #pragma unroll
  for (int d = 0; d < 3; ++d) {
    #pragma unroll
    for (int s = 0; s < 4; ++s) {
      v2f b = *(const v2f*)(W + d * (COUT * CIN) + m * CIN + 4 * s + 2 * half);
      c[d] = __builtin_amdgcn_wmma_f32_16x16x4_f32(
          /*neg_a=*/false, a[s], /*neg_b=*/false, b,
          /*c_mod=*/(short)0, c[d], /*reuse_a=*/false, /*reuse_b=*/false);
    }
  }

  // D layout: VGPR r, lane L -> M = r + 8*(L/16), N = L%16
  #pragma unroll
  for (int d = 0; d < 3; ++d) {
    #pragma unroll
    for (int r = 0; r < 8; ++r)
      G[(tile * 16 + r + 8 * half) * GCOLS + d * COUT + m] = c[d][r];
  }
}

// ---------------------------------------------------------------------------
// Phase 2: out[p,o] = sum_k sum_d radii[p,k,d] * G[nbr[p,k], d*16+o]
// Edges are point-major -> segment_sum is a private per-point reduction,
// no atomics. Lane = (point, channel); 16 lanes read contiguous 64B G rows
// that stay resident in the 192MB L2 (G is only 7.7MB total).
// ---------------------------------------------------------------------------
__global__ __launch_bounds__(256) void pconv_gather_reduce(
    const float* __restrict__ G, const float* __restrict__ radii,
    const int* __restrict__ bs, float* __restrict__ out)
{
  const int gid = blockIdx.x * 256 + threadIdx.x;   // NPTS*COUT threads exactly
  const int p   = gid >> 4;
  const int o   = gid & 15;

  const int*   nb = bs    + p * (1 + DEG) + 1;      // skip count column
  const float* rr = radii + (size_t)p * DEG * 3;

  float acc = 0.0f;
  #pragma unroll 4
  for (int k = 0; k < DEG; ++k) {
    const int    nbr = nb[k];
    const float  r0  = rr[3 * k + 0];
    const float  r1  = rr[3 * k + 1];
    const float  r2  = rr[3 * k + 2];
    const float* g   = G + (size_t)nbr * GCOLS;
    acc = fmaf(r0, g[o],            acc);
    acc = fmaf(r1, g[COUT + o],     acc);
    acc = fmaf(r2, g[2 * COUT + o], acc);
  }
  out[p * COUT + o] = acc;
}

extern "C" void kernel_launch(void* const* d_in, const int* in_sizes, int n_in,
                              void* d_out, int out_size, void* d_ws, size_t ws_size,
                              hipStream_t stream) {
  const float* F     = (const float*)d_in[0];   // features [P,16]
  const float* radii = (const float*)d_in[1];   // [E,3]
  const float* W     = (const float*)d_in[2];   // [3,16,16]
  const int*   bs    = (const int*)  d_in[3];   // [P,17]
  float*       out   = (float*)d_out;
  float*       G     = (float*)d_ws;            // [P,48] = 7.68 MB scratch

  // Phase 1: 2500 tiles / 4 waves per block
  pconv_point_transform<<<625, 128, 0, stream>>>(F, W, G);
  // Phase 2: NPTS*COUT = 640000 threads = 2500 blocks x 256
  pconv_gather_reduce<<<2500, 256, 0, stream>>>(G, radii, bs, out);
}